// SpearmanWithBCE_65146063946040
// MI455X (gfx1250) — compile-verified
//
#include <hip/hip_runtime.h>
#include <hip/hip_bf16.h>

// ---------------------------------------------------------------------------
// Fused SpearmanWithBCE for MI455X (gfx1250, wave32).
// Single workgroup, everything resident in LDS (96KB of the WGP's 320KB).
//   Phase 0: async-stage y -> LDS (GLOBAL_LOAD_ASYNC_TO_LDS_B128, ASYNCcnt)
//            while sorting y_hat; bitonic argsort descending; PAV isotonic
//            (non-increasing) on s - [n..1]; scatter primal via permutation.
//   Phase 1: same for y, keys sourced from the async-staged LDS copy.
//   Phase 2: six f32 reductions via V_WMMA_F32_16X16X4_F32 ones-matmul
//            (64 elems folded per WMMA, exact f32 accumulation), prefetched
//            global re-reads for the BCE terms.
// ---------------------------------------------------------------------------

typedef __attribute__((ext_vector_type(2))) float v2f;
typedef __attribute__((ext_vector_type(8))) float v8f;
typedef __attribute__((address_space(3))) float lds_f;

#define NN 4096
#define NT 256
#define NWAVE 8
#define WAVE_ELEMS (NN / NWAVE)   // 512 elements per wave in reduction
#define CHUNKS (WAVE_ELEMS / 64)  // 8 WMMA issues per channel per wave

__launch_bounds__(NT)
__global__ void spearman_bce_fused(const float* __restrict__ yhat,
                                   const float* __restrict__ ytgt,
                                   float* __restrict__ out) {
  __shared__ float s_key[NN];    // sort keys (sorted s after bitonic)
  __shared__ int   s_idx[NN];    // permutation (original indices)
  __shared__ float s_rankP[NN];  // soft ranks of y_hat (original order)
  __shared__ float s_rankT[NN];  // pass 0: async staging of y; then ranks of y
  __shared__ float s_bsum[NN];   // PAV block sums (stack)
  __shared__ int   s_bcnt[NN];   // PAV block counts (stack)
  __shared__ float s_part[NWAVE * 8];

  const int tid = threadIdx.x;

  // ---- kick off async global->LDS staging of y into s_rankT --------------
  // 4096 floats = 4 x b128 per thread; latency hidden behind pass-0 sort.
  // s_rankT is dead storage until pass 1 consumes it, then it is overwritten
  // by pass 1's rank output (after the staged copy is no longer needed).
#pragma unroll
  for (int v = 0; v < 4; ++v) {
    const int e = v * 1024 + tid * 4;           // 16B-aligned float index
    const float* g = ytgt + e;
    lds_f* l = (lds_f*)&s_rankT[e];             // raw LDS offset operand
    asm volatile("global_load_async_to_lds_b128 %0, %1, off"
                 :: "v"(l), "v"(g) : "memory");
  }

  // ---------------- soft_rank for both vectors ----------------
  for (int pass = 0; pass < 2; ++pass) {
    float* dst = (pass == 0) ? s_rankP : s_rankT;

    if (pass == 0) {
      for (int i = tid; i < NN; i += NT) { s_key[i] = yhat[i]; s_idx[i] = i; }
    } else {
      // drain the async staging before anyone reads it
      asm volatile("s_wait_asynccnt 0x0" ::: "memory");
      __syncthreads();
      for (int i = tid; i < NN; i += NT) { s_key[i] = s_rankT[i]; s_idx[i] = i; }
    }
    __syncthreads();

    // Bitonic sort, descending overall. 78 passes, 2048 pairs/pass (8/thread avg).
    for (int k = 2; k <= NN; k <<= 1) {
      for (int j = k >> 1; j > 0; j >>= 1) {
        for (int i = tid; i < NN; i += NT) {
          const int ixj = i ^ j;
          if (ixj > i) {
            const float a = s_key[i];
            const float b = s_key[ixj];
            const bool descend = ((i & k) == 0);
            const bool sw = descend ? (a < b) : (a > b);
            if (sw) {
              s_key[i] = b; s_key[ixj] = a;
              const int ta = s_idx[i]; s_idx[i] = s_idx[ixj]; s_idx[ixj] = ta;
            }
          }
        }
        __syncthreads();
      }
    }

    // PAV: isotonic NON-INCREASING fit of z[i] = s[i] - w[i], w = [n..1].
    // O(n) stack algorithm, one lane, LDS-resident stack. Mean comparison via
    // cross-multiplication (counts > 0) to avoid divides in the hot loop.
    if (tid == 0) {
      int nb = 0;
      for (int i = 0; i < NN; ++i) {
        s_bsum[nb] = s_key[i] - (float)(NN - i);
        s_bcnt[nb] = 1;
        ++nb;
        while (nb >= 2 &&
               s_bsum[nb - 1] * (float)s_bcnt[nb - 2] >
               s_bsum[nb - 2] * (float)s_bcnt[nb - 1]) {
          s_bsum[nb - 2] += s_bsum[nb - 1];
          s_bcnt[nb - 2] += s_bcnt[nb - 1];
          --nb;
        }
      }
      // Expand blocks: primal = s - dual, scatter through permutation.
      int pos = 0;
      for (int b = 0; b < nb; ++b) {
        const float mean = s_bsum[b] / (float)s_bcnt[b];
        const int cnt = s_bcnt[b];
        for (int r = 0; r < cnt; ++r, ++pos) {
          dst[s_idx[pos]] = s_key[pos] - mean;
        }
      }
    }
    __syncthreads();
  }

  // ---- prefetch the BCE re-reads (evicted from WGP$ during the sorts) ----
  if (tid < 128) {
    __builtin_prefetch(yhat + tid * 32, 0, 3);   // 128 lines x 128B = 16KB
  } else {
    __builtin_prefetch(ytgt + (tid - 128) * 32, 0, 3);
  }

  // ---------------- reductions on the matrix pipe ----------------
  // D = ones(16x4) * B(4x16) + C  =>  every D column accumulates a disjoint
  // 4-subset of the 64 values fed as B; all 16 D rows are identical, so any
  // lane->B-slot bijection is valid for a total sum. Chained C gives exact
  // f32 accumulation across chunks. EXEC is all-ones (full 256-thread block).
  const int wave = tid >> 5;
  const int lane = tid & 31;
  v2f onesA; onesA[0] = 1.0f; onesA[1] = 1.0f;
  v8f accP = {}, accPP = {}, accT = {}, accTT = {}, accPT = {}, accB = {};

  const int base = wave * WAVE_ELEMS;
  for (int c = 0; c < CHUNKS; ++c) {
    const int e0 = base + c * 64 + lane * 2;
    const int e1 = e0 + 1;
    const float p0 = s_rankP[e0], p1 = s_rankP[e1];
    const float t0 = s_rankT[e0], t1 = s_rankT[e1];
    const float x0 = yhat[e0],    x1 = yhat[e1];
    const float g0 = ytgt[e0],    g1 = ytgt[e1];
    // logaddexp(0,x) - x*y  (stable form)
    const float b0 = log1pf(expf(-fabsf(x0))) + fmaxf(x0, 0.0f) - x0 * g0;
    const float b1 = log1pf(expf(-fabsf(x1))) + fmaxf(x1, 0.0f) - x1 * g1;

    v2f vp  = {p0,       p1};
    v2f vpp = {p0 * p0,  p1 * p1};
    v2f vt  = {t0,       t1};
    v2f vtt = {t0 * t0,  t1 * t1};
    v2f vpt = {p0 * t0,  p1 * t1};
    v2f vb  = {b0,       b1};

    accP  = __builtin_amdgcn_wmma_f32_16x16x4_f32(false, onesA, false, vp,  (short)0, accP,  false, false);
    accPP = __builtin_amdgcn_wmma_f32_16x16x4_f32(false, onesA, false, vpp, (short)0, accPP, false, false);
    accT  = __builtin_amdgcn_wmma_f32_16x16x4_f32(false, onesA, false, vt,  (short)0, accT,  false, false);
    accTT = __builtin_amdgcn_wmma_f32_16x16x4_f32(false, onesA, false, vtt, (short)0, accTT, false, false);
    accPT = __builtin_amdgcn_wmma_f32_16x16x4_f32(false, onesA, false, vpt, (short)0, accPT, false, false);
    accB  = __builtin_amdgcn_wmma_f32_16x16x4_f32(false, onesA, false, vb,  (short)0, accB,  false, false);
  }

  // D row 0 lives in acc[0]: lanes 0..15 hold columns 0..15 (lanes 16..31 are
  // duplicate rows). Fold the 16 column partials with a 4-step xor shuffle.
  float r[6];
  r[0] = accP[0]; r[1] = accPP[0]; r[2] = accT[0];
  r[3] = accTT[0]; r[4] = accPT[0]; r[5] = accB[0];
#pragma unroll
  for (int ch = 0; ch < 6; ++ch) {
    float v = r[ch];
    v += __shfl_xor(v, 1, 32);
    v += __shfl_xor(v, 2, 32);
    v += __shfl_xor(v, 4, 32);
    v += __shfl_xor(v, 8, 32);
    if (lane == 0) s_part[wave * 8 + ch] = v;
  }
  __syncthreads();

  if (tid == 0) {
    float S[6] = {0.f, 0.f, 0.f, 0.f, 0.f, 0.f};
    for (int w = 0; w < NWAVE; ++w)
      for (int ch = 0; ch < 6; ++ch) S[ch] += s_part[w * 8 + ch];
    const float n = (float)NN;
    const float Sp = S[0], Spp = S[1], St = S[2], Stt = S[3], Spt = S[4], Sb = S[5];
    const float cov = Spt - Sp * St / n;
    const float vp  = Spp - Sp * Sp / n;
    const float vt  = Stt - St * St / n;
    const float spearman = cov * rsqrtf(vp) * rsqrtf(vt);
    const float bce = Sb / n;
    out[0] = 1.0f - spearman + bce;  // W1*(1-spearman) + W2*bce, W1=W2=1
  }
}

extern "C" void kernel_launch(void* const* d_in, const int* in_sizes, int n_in,
                              void* d_out, int out_size, void* d_ws, size_t ws_size,
                              hipStream_t stream) {
  (void)in_sizes; (void)n_in; (void)out_size; (void)d_ws; (void)ws_size;
  const float* y_hat = (const float*)d_in[0];
  const float* y     = (const float*)d_in[1];
  float* out         = (float*)d_out;
  spearman_bce_fused<<<dim3(1), dim3(NT), 0, stream>>>(y_hat, y, out);
}